// C3DLoss_29772713296415
// MI455X (gfx1250) — compile-verified
//
#include <hip/hip_runtime.h>
#include <hip/hip_bf16.h>
#include <stdint.h>

// ---------------- problem constants (from reference) ----------------
#define Wd 1216
#define Hd 352
#define Bd 8
#define TW 32
#define TH 8
#define HALO 2
#define MAXCW (TW + 2*HALO)   // 36  (fixed LDS tile pitch)
#define MAXCH (TH + 2*HALO)   // 12
#define NPIX  (MAXCH * MAXCW) // 432

#define FXi  (1.0f / 720.0f)
#define CXc 608.0f
#define CYc 176.0f
// ELL_XYZ = 0.05+0.5*0.1 = 0.1 ; ELL_HSV = 0.1  ->  1/(2*ell^2) = 50 (both)
#define INV2E 50.0f
#define CROSS_W 1.0f

typedef unsigned int u32x4 __attribute__((ext_vector_type(4)));
typedef int          i32x8 __attribute__((ext_vector_type(8)));
typedef int          i32x4 __attribute__((ext_vector_type(4)));

#if __has_builtin(__builtin_amdgcn_tensor_load_to_lds)
#define HAVE_TDM 1
#else
#define HAVE_TDM 0
#endif

// Low 32 bits of a generic pointer to __shared__ = LDS byte offset (flat aperture).
template <typename T>
__device__ __forceinline__ unsigned ldsAddrOf(T* p) {
    return (unsigned)(uintptr_t)(void*)p;
}

#if HAVE_TDM
// TDM 2D tile load: fixed 12 rows x 36 elements, global row stride Wd,
// packed contiguously in LDS (pitch 36).  Descriptor is compile-time constant
// except the two addresses.  D# layout per CDNA5 ISA ch.8.
__device__ __forceinline__ void tdm_load_tile(unsigned lds_byte_addr,
                                              const void* gptr,
                                              int data_size_code /*0:1B 2:4B*/) {
    unsigned long long ga = (unsigned long long)(uintptr_t)gptr;
    u32x4 g0 = {
        1u,                                             // count=1 (valid user D#)
        lds_byte_addr,                                  // lds_addr [63:32]
        (unsigned)(ga & 0xFFFFFFFFu),                   // global_addr[31:0]
        (unsigned)((ga >> 32) & 0x01FFFFFFu) | (2u<<30) // global_addr[56:32], type=2
    };
    i32x8 g1 = {
        (int)((unsigned)(data_size_code & 3) << 16),    // wg_mask=0, data_size
        (int)(((unsigned)MAXCW) << 16),                 // tensor_dim0 = 36
        (int)(((unsigned)MAXCH) << 16),                 // tensor_dim1 = 12
        (int)(((unsigned)MAXCW) << 16),                 // tile_dim0 = 36
        (int)((unsigned)MAXCH),                         // tile_dim1 = 12, tile_dim2=0
        (int)((unsigned)Wd),                            // tensor_dim0_stride = 1216
        0, 0
    };
    i32x4 z4 = {0, 0, 0, 0};
    i32x8 z8 = {0, 0, 0, 0, 0, 0, 0, 0};
    __builtin_amdgcn_tensor_load_to_lds(g0, g1, z4, z4, z8, 0);
}
#endif

// ------------------------------- main kernel --------------------------------
__global__ __launch_bounds__(256) void c3d_main(
        const float* __restrict__ rgb,
        const float* __restrict__ depth,
        const float* __restrict__ depth_gt,
        const unsigned char* __restrict__ mpB,   // depth_mask    (numpy bool, 1B)
        const unsigned char* __restrict__ mgB,   // depth_gt_mask (numpy bool, 1B)
        const float* __restrict__ T21,           // ws+16: 8 x (R[9], t[3])
        float* __restrict__ acc)                 // ws[0]=same ws[1]=cross ws[2]=n_gt
{
    __shared__ float sD[NPIX];
    __shared__ float sR[NPIX];
    __shared__ float sG[NPIX];
    __shared__ float sB[NPIX];
    __shared__ __attribute__((aligned(16))) unsigned char sM[NPIX];
    __shared__ float sAcc[3];

    const int b  = blockIdx.z;
    const int x0 = blockIdx.x * TW;
    const int y0 = blockIdx.y * TH;
    // Clamped full-size window: always 12x36, always fully inside the image,
    // so every LDS access below is valid; only *image* validity varies per tap.
    int cx0 = x0 - HALO; cx0 = cx0 < 0 ? 0 : (cx0 > Wd - MAXCW ? Wd - MAXCW : cx0);
    int cy0 = y0 - HALO; cy0 = cy0 < 0 ? 0 : (cy0 > Hd - MAXCH ? Hd - MAXCH : cy0);
    const int tid = threadIdx.x;

    const size_t plane = (size_t)Hd * Wd;
    const float* db = depth + (size_t)b * plane;
    const float* rb = rgb   + (size_t)b * 3 * plane;
    const size_t go = (size_t)cy0 * Wd + cx0;

#if HAVE_TDM
    if (tid < 32) {  // single wave issues the DMA set (TDM ignores EXEC)
        tdm_load_tile(ldsAddrOf(sD), db + go,             2);
        tdm_load_tile(ldsAddrOf(sR), rb + go,             2);
        tdm_load_tile(ldsAddrOf(sG), rb + plane + go,     2);
        tdm_load_tile(ldsAddrOf(sB), rb + 2 * plane + go, 2);
        tdm_load_tile(ldsAddrOf(sM), mpB + (size_t)b * plane + go, 0);
        __builtin_amdgcn_s_wait_tensorcnt(0);
    }
#else
    for (int i = tid; i < NPIX; i += 256) {
        int ry = i / MAXCW, rx = i - ry * MAXCW;
        size_t gi = (size_t)(cy0 + ry) * Wd + (cx0 + rx);
        sD[i] = db[gi];
        sR[i] = rb[gi];
        sG[i] = rb[plane + gi];
        sB[i] = rb[2 * plane + gi];
        sM[i] = mpB[(size_t)b * plane + gi];
    }
#endif
    if (tid < 3) sAcc[tid] = 0.0f;
    __syncthreads();

    // ---- per-thread center pixel (tiles divide H,W exactly: 1216/32, 352/8)
    const int tx = tid & (TW - 1);
    const int ty = tid >> 5;
    const int x = x0 + tx, y = y0 + ty;
    const float uc = (x - CXc) * FXi;
    const float vc = (y - CYc) * FXi;
    const size_t pidx = (size_t)y * Wd + x;
    const int li = (y - cy0) * MAXCW + (x - cx0);   // constant-pitch base index

    // reference-frame (same term): xyz_g = xy1 * depth_gt ; rgb_ref from LDS
    const float dgt = depth_gt[(size_t)b * plane + pidx];
    const float mg  = (float)mgB[(size_t)b * plane + pidx];
    const float rr = sR[li], rg = sG[li], rbl = sB[li];
    const float gx = uc * dgt, gy = vc * dgt, gz = dgt;

    // cross-frame reference: batch tgt = b^1 (SEQ_N=2 pair swap)
    const int bt = b ^ 1;
    const float dgt2 = depth_gt[(size_t)bt * plane + pidx];
    const float mg2  = (float)mgB[(size_t)bt * plane + pidx];
    const float rr2  = rgb[(size_t)bt * 3 * plane + pidx];
    const float rg2  = rgb[(size_t)bt * 3 * plane + plane + pidx];
    const float rb2  = rgb[(size_t)bt * 3 * plane + 2 * plane + pidx];

    float R[9], t3[3];
    #pragma unroll
    for (int i = 0; i < 9; ++i) R[i] = T21[b * 12 + i];
    #pragma unroll
    for (int i = 0; i < 3; ++i) t3[i] = T21[b * 12 + 9 + i];
    // fold (g2 - t) per thread:  diff = o - dq*w,  w = R*(uq,vq,1)
    const float o0 = uc * dgt2 - t3[0];
    const float o1 = vc * dgt2 - t3[1];
    const float o2 = dgt2      - t3[2];
    const float g2z = dgt2;   // tzp > 0  <=>  fz0 < g2z  (since fz0 = g2z - tzp)

    float sacc = 0.0f, cacc = 0.0f;
    #pragma unroll
    for (int dy = -HALO; dy <= HALO; ++dy) {
        const int yq = y + dy;
        const bool yok = (unsigned)yq < (unsigned)Hd;
        const float vq = vc + (float)dy * FXi;
        // row-hoisted rotation partials: w = uq*Rcol0 + (vq*Rcol1 + Rcol2)
        const float hx = fmaf(R[1], vq, R[2]);
        const float hy = fmaf(R[4], vq, R[5]);
        const float hz = fmaf(R[7], vq, R[8]);
        const int rbase = li + dy * MAXCW;          // literal row offset
        #pragma unroll
        for (int dx = -HALO; dx <= HALO; ++dx) {
            const int xq = x + dx;
            const int qi = rbase + dx;              // literal ds offsets, always valid
            // zero-padding of the reference: kill contribution via mq, no branch
            const bool ok = yok && ((unsigned)xq < (unsigned)Wd);
            float mq = (float)sM[qi];
            mq = ok ? mq : 0.0f;
            const float dq = sD[qi];
            const float uq = uc + (float)dx * FXi;
            const float cr = sR[qi], cg = sG[qi], cb = sB[qi];
            // same-frame term: 6 squared diffs, one exp
            const float qx = uq * dq, qy = vq * dq;
            float ex = gx - qx, ey = gy - qy, ez = gz - dq;
            float e0 = rr - cr, e1 = rg - cg, e2 = rbl - cb;
            float d2 = ex*ex + ey*ey + ez*ez + e0*e0 + e1*e1 + e2*e2;
            sacc = fmaf(__expf(-INV2E * d2), mq, sacc);
            // cross-frame term: diff = o - dq*w ; positivity via fz0 < g2z
            const float wx = fmaf(R[0], uq, hx);
            const float wy = fmaf(R[3], uq, hy);
            const float wz = fmaf(R[6], uq, hz);
            float fx0 = fmaf(-dq, wx, o0);
            float fy0 = fmaf(-dq, wy, o1);
            float fz0 = fmaf(-dq, wz, o2);
            const float pos = (fz0 < g2z) ? mq : 0.0f;
            float c0 = rr2 - cr, c1 = rg2 - cg, c2 = rb2 - cb;
            float d2c = fx0*fx0 + fy0*fy0 + fz0*fz0 + c0*c0 + c1*c1 + c2*c2;
            cacc = fmaf(__expf(-INV2E * d2c), pos, cacc);
        }
    }
    sacc *= mg;    // hoisted center-mask multiply
    cacc *= mg2;
    float ng = mg; // n_gt contribution

    // wave32 tree reduction, then LDS atomics, then one global atomic set/block
    #pragma unroll
    for (int off = 16; off > 0; off >>= 1) {
        sacc += __shfl_xor(sacc, off, 32);
        cacc += __shfl_xor(cacc, off, 32);
        ng   += __shfl_xor(ng,   off, 32);
    }
    if ((tid & 31) == 0) {
        atomicAdd(&sAcc[0], sacc);
        atomicAdd(&sAcc[1], cacc);
        atomicAdd(&sAcc[2], ng);
    }
    __syncthreads();
    if (tid == 0) {
        atomicAdd(&acc[0], sAcc[0]);
        atomicAdd(&acc[1], sAcc[1]);
        atomicAdd(&acc[2], sAcc[2]);
    }
}

// -------- setup: zero accumulators; build T21 (rigid inverse) per batch -----
__global__ void c3d_setup(const float* __restrict__ Ts, float* __restrict__ ws) {
    int b = threadIdx.x;
    if (b < 3) ws[b] = 0.0f;
    if (b < Bd) {
        int t = b ^ 1;  // tgt for SEQ_N=2: pair swap [1,0,3,2,...]
        const float* T2 = Ts + b * 16;   // Ts
        const float* T1 = Ts + t * 16;   // Ts[tgt]
        float* o = ws + 16 + b * 12;
        float d0 = T2[3]  - T1[3];
        float d1 = T2[7]  - T1[7];
        float d2 = T2[11] - T1[11];
        #pragma unroll
        for (int i = 0; i < 3; ++i) {
            // row i of R21 = (col i of R1) dot (cols of R2);  t21 = R1^T (t2-t1)
            #pragma unroll
            for (int j = 0; j < 3; ++j)
                o[i*3 + j] = T1[0*4+i]*T2[0*4+j] + T1[1*4+i]*T2[1*4+j] + T1[2*4+i]*T2[2*4+j];
            o[9 + i] = T1[0*4+i]*d0 + T1[1*4+i]*d1 + T1[2*4+i]*d2;
        }
    }
}

// ------------------------------ finalize ------------------------------------
__global__ void c3d_final(const float* __restrict__ acc, float* __restrict__ out) {
    float ng = fmaxf(acc[2], 1.0f);
    out[0] = -(acc[0] + CROSS_W * acc[1]) / ng;
}

// ------------------------------ launch --------------------------------------
extern "C" void kernel_launch(void* const* d_in, const int* in_sizes, int n_in,
                              void* d_out, int out_size, void* d_ws, size_t ws_size,
                              hipStream_t stream) {
    const float*         rgb      = (const float*)d_in[0];
    const float*         depth    = (const float*)d_in[1];
    const float*         depth_gt = (const float*)d_in[2];
    const unsigned char* mp       = (const unsigned char*)d_in[3]; // bool mask
    const unsigned char* mg       = (const unsigned char*)d_in[4]; // bool mask
    // d_in[5] = xy1_grid: recomputed analytically on-chip, not loaded
    const float*         Ts       = (const float*)d_in[6];
    float* ws = (float*)d_ws;

    c3d_setup<<<1, 32, 0, stream>>>(Ts, ws);
    dim3 grid(Wd / TW, Hd / TH, Bd);
    c3d_main<<<grid, 256, 0, stream>>>(rgb, depth, depth_gt, mp, mg, ws + 16, ws);
    c3d_final<<<1, 1, 0, stream>>>(ws, (float*)d_out);
}